// RNNModel_35442070126667
// MI455X (gfx1250) — compile-verified
//
#include <hip/hip_runtime.h>
#include <hip/hip_bf16.h>
#include <math.h>

#define B_ 64
#define S_ 512
#define V_ 2048
#define H_ 1024

typedef float v2f __attribute__((ext_vector_type(2)));
typedef float v8f __attribute__((ext_vector_type(8)));

// ---------------------------------------------------------------------------
// Kernel 1: transpose W_ih [H,V] -> W_ihT [V,H] (LDS tiled, conflict-padded)
// ---------------------------------------------------------------------------
__global__ __launch_bounds__(256) void k_transpose_wih(const float* __restrict__ W_ih,
                                                       float* __restrict__ W_ihT) {
    __shared__ float tile[16][17];
    const int tx = threadIdx.x & 15;
    const int ty = threadIdx.x >> 4;
    const int v0 = blockIdx.x * 16;
    const int h0 = blockIdx.y * 16;
    tile[ty][tx] = W_ih[(size_t)(h0 + ty) * V_ + (v0 + tx)];
    __syncthreads();
    W_ihT[(size_t)(v0 + ty) * H_ + (h0 + tx)] = tile[tx][ty];
}

// ---------------------------------------------------------------------------
// Kernel 2: one recurrence step.
//   Hout[b,h] = tanh( Hprev @ W_hh^T + W_ihT[tok[b,s]] + b_ih + b_hh )
// One wave per 16x16 output tile; K-loop of v_wmma_f32_16x16x4_f32.
// Grid: 64 blocks x 128 threads (4 waves) == 256 tiles exactly (no masking,
// EXEC stays all-ones as WMMA requires).
// ---------------------------------------------------------------------------
__global__ __launch_bounds__(128) void k_rnn_step(const float* __restrict__ Hprev, // [B,H]
                                                  const float* __restrict__ W_hh,  // [H,H] (N x K)
                                                  const float* __restrict__ W_ihT, // [V,H]
                                                  const int*   __restrict__ inp,   // [B,S]
                                                  const float* __restrict__ b_ih,
                                                  const float* __restrict__ b_hh,
                                                  float* __restrict__ Hout,        // [B,H]
                                                  int s) {
    const int wave = threadIdx.x >> 5;
    const int lane = threadIdx.x & 31;
    const int t    = blockIdx.x * 4 + wave;   // tile id, 0..255
    const int tm   = t >> 6;                  // 0..3   (B/16)
    const int tn   = t & 63;                  // 0..63  (H/16)
    const int hl   = lane >> 4;               // lane half selects K pair
    const int lid  = lane & 15;

    // A: Hprev tile rows (M = batch), B: W_hh rows (N = hidden out), both along K
    const float* __restrict__ Arow = Hprev + (size_t)(tm * 16 + lid) * H_;
    const float* __restrict__ Brow = W_hh  + (size_t)(tn * 16 + lid) * H_;

    v8f acc = {};
    for (int k = 0; k < H_; k += 4) {
        const int kk = k + hl * 2;
        v2f a = *(const v2f*)(Arow + kk);
        v2f b = *(const v2f*)(Brow + kk);
        acc = __builtin_amdgcn_wmma_f32_16x16x4_f32(false, a, false, b,
                                                    (short)0, acc, false, false);
    }

    // Epilogue: fused embedding gather + biases + tanh
    const int   n    = tn * 16 + lid;
    const float bias = b_ih[n] + b_hh[n];
#pragma unroll
    for (int r = 0; r < 8; ++r) {
        const int m   = tm * 16 + r + hl * 8;      // batch row per C/D layout
        const int tok = inp[(size_t)m * S_ + s];
        const float v = acc[r] + W_ihT[(size_t)tok * H_ + n] + bias;
        Hout[(size_t)m * H_ + n] = tanhf(v);
    }
}

// ---------------------------------------------------------------------------
// Kernel 3: readout GEMM  Out[SB, V] = Y[SB, H] @ W_out^T + b_out
// Each wave computes a 64x64 block (4x4 WMMA tiles): 8 fragment loads feed
// 16 wmma per k-step -> 16 FLOP/byte from cache (2x the 2x2 version).
// Grid: 4096 blocks x 128 threads = 16384 waves = (32768/64)*(2048/64) exact.
// ---------------------------------------------------------------------------
__global__ __launch_bounds__(128) void k_out_gemm(const float* __restrict__ Y,     // [SB,H]
                                                  const float* __restrict__ W_out, // [V,H]
                                                  const float* __restrict__ b_out, // [V]
                                                  float* __restrict__ Out) {       // [SB,V]
    const int wave = threadIdx.x >> 5;
    const int lane = threadIdx.x & 31;
    const int wt   = blockIdx.x * 4 + wave;
    const int wm   = wt >> 5;   // 0..511  (32768/64)
    const int wn   = wt & 31;   // 0..31   (2048/64)
    const int hl   = lane >> 4;
    const int lid  = lane & 15;

    const float* __restrict__ Ap[4];
    const float* __restrict__ Bp[4];
#pragma unroll
    for (int i = 0; i < 4; ++i) {
        Ap[i] = Y     + (size_t)(wm * 64 + i * 16 + lid) * H_;
        Bp[i] = W_out + (size_t)(wn * 64 + i * 16 + lid) * H_;
    }

    v8f acc[4][4];
#pragma unroll
    for (int i = 0; i < 4; ++i)
#pragma unroll
        for (int j = 0; j < 4; ++j) acc[i][j] = (v8f){};

    for (int k = 0; k < H_; k += 4) {
        const int kk = k + hl * 2;
        v2f a[4], b[4];
#pragma unroll
        for (int i = 0; i < 4; ++i) a[i] = *(const v2f*)(Ap[i] + kk);
#pragma unroll
        for (int j = 0; j < 4; ++j) b[j] = *(const v2f*)(Bp[j] + kk);
#pragma unroll
        for (int i = 0; i < 4; ++i)
#pragma unroll
            for (int j = 0; j < 4; ++j)
                acc[i][j] = __builtin_amdgcn_wmma_f32_16x16x4_f32(
                    false, a[i], false, b[j], (short)0, acc[i][j], false, false);
    }

#pragma unroll
    for (int j = 0; j < 4; ++j) {
        const int   n  = wn * 64 + j * 16 + lid;
        const float bb = b_out[n];
#pragma unroll
        for (int i = 0; i < 4; ++i) {
#pragma unroll
            for (int r = 0; r < 8; ++r) {
                const size_t m = (size_t)(wm * 64 + i * 16 + r + hl * 8);
                Out[m * V_ + n] = acc[i][j][r] + bb;
            }
        }
    }
}

// ---------------------------------------------------------------------------
// Kernel 4: append h_last = Y[S-1] to the tail of d_out
// ---------------------------------------------------------------------------
__global__ void k_copy(const float* __restrict__ src, float* __restrict__ dst, int n) {
    int i = blockIdx.x * blockDim.x + threadIdx.x;
    if (i < n) dst[i] = src[i];
}

// ---------------------------------------------------------------------------
extern "C" void kernel_launch(void* const* d_in, const int* in_sizes, int n_in,
                              void* d_out, int out_size, void* d_ws, size_t ws_size,
                              hipStream_t stream) {
    const int*   inp   = (const int*)  d_in[0];  // [B,S]
    const float* state = (const float*)d_in[1];  // [1,B,H]
    const float* W_ih  = (const float*)d_in[2];  // [H,V]
    const float* W_hh  = (const float*)d_in[3];  // [H,H]
    const float* b_ih  = (const float*)d_in[4];  // [H]
    const float* b_hh  = (const float*)d_in[5];  // [H]
    const float* W_out = (const float*)d_in[6];  // [V,H]
    const float* b_out = (const float*)d_in[7];  // [V]
    float*       out   = (float*)d_out;          // [S*B,V] then [B,H]

    float* W_ihT = (float*)d_ws;                       // V*H floats (8 MB)
    float* Y     = W_ihT + (size_t)V_ * H_;            // S*B*H floats (134 MB)

    // 1) transpose embedding weights for coalesced row gather
    dim3 tg(V_ / 16, H_ / 16);
    k_transpose_wih<<<tg, 256, 0, stream>>>(W_ih, W_ihT);

    // 2) sequential recurrence: one kernel per step (captured into the graph)
    for (int s = 0; s < S_; ++s) {
        const float* Hprev = (s == 0) ? state : (Y + (size_t)(s - 1) * B_ * H_);
        float*       Hout  = Y + (size_t)s * B_ * H_;
        k_rnn_step<<<64, 128, 0, stream>>>(Hprev, W_hh, W_ihT, inp, b_ih, b_hh, Hout, s);
    }

    // 3) readout GEMM over all timesteps at once (64x64 per wave, 4x4 WMMA tiles)
    k_out_gemm<<<(S_ * B_ / 64) * (V_ / 64) / 4, 128, 0, stream>>>(Y, W_out, b_out, out);

    // 4) final hidden state
    k_copy<<<(B_ * H_ + 255) / 256, 256, 0, stream>>>(Y + (size_t)(S_ - 1) * B_ * H_,
                                                      out + (size_t)S_ * B_ * V_, B_ * H_);
}